// GIN_56727928046274
// MI455X (gfx1250) — compile-verified
//
#include <hip/hip_runtime.h>
#include <hip/hip_bf16.h>

typedef float v2f __attribute__((ext_vector_type(2)));
typedef float v8f __attribute__((ext_vector_type(8)));

#define IN_F 64
#define OUT_F 128
#define BN_EPS_F 1e-5f

// ---------------------------------------------------------------------------
// ws layout (floats):
//   h     : [N][64]       (x + segment_sum)
//   z     : [N][128]      (h @ W1 + b1)
//   stats : sum[128], sumsq[128], scale[128], shift[128]
// ---------------------------------------------------------------------------

__global__ void gin_init(const float* __restrict__ x, float* __restrict__ h,
                         float* __restrict__ stats, int total4) {
  int i = blockIdx.x * blockDim.x + threadIdx.x;
  if (i < total4) ((float4*)h)[i] = ((const float4*)x)[i];
  if (i < 256) stats[i] = 0.0f;  // zero sum/sumsq every call (ws is poisoned)
}

// one 32-lane group per edge; 2 features per lane; f32 scatter atomics
__global__ void gin_scatter(const float* __restrict__ x,
                            const long long* __restrict__ src,
                            const long long* __restrict__ dst,
                            float* __restrict__ h, int E) {
  unsigned idx = blockIdx.x * 256u + threadIdx.x;
  int e = (int)(idx >> 5);
  int lane = (int)(idx & 31u);
  if (e < E) {
    long long s = src[e];
    long long d = dst[e];
    float2 v = ((const float2*)(x + s * IN_F))[lane];
    float* hp = h + d * IN_F + lane * 2;
    atomicAdd(hp, v.x);
    atomicAdd(hp + 1, v.y);
  }
}

// Stage 64 rows x 128 cols of W (row-major) into K-pair-interleaved LDS:
// ldsP[k/2][col] = { W[k][col], W[k+1][col] }.  Pair-row stride 144 float2
// (1152 B) puts lane-halves (pair-rows p and p+1) on disjoint bank halves.
#define PAIR_STRIDE 144

__device__ __forceinline__ void stage_w_pairs(v2f (*ldsP)[PAIR_STRIDE],
                                              const float* __restrict__ Wsrc,
                                              int tid) {
  const float4* w4 = (const float4*)Wsrc;
  for (int i = tid; i < 32 * 32; i += 256) {
    int pr = i >> 5;          // pair-row 0..31
    int c4 = (i & 31) * 4;    // column group
    float4 a0 = w4[(2 * pr) * 32 + (c4 >> 2)];
    float4 a1 = w4[(2 * pr + 1) * 32 + (c4 >> 2)];
    ldsP[pr][c4 + 0] = (v2f){a0.x, a1.x};
    ldsP[pr][c4 + 1] = (v2f){a0.y, a1.y};
    ldsP[pr][c4 + 2] = (v2f){a0.z, a1.z};
    ldsP[pr][c4 + 3] = (v2f){a0.w, a1.w};
  }
}

// z = h @ W1 + b1 (K=64) with fused per-column BN statistics (sum, sumsq)
__global__ __launch_bounds__(256) void gin_gemm1(
    const float* __restrict__ h, const float* __restrict__ W1,
    const float* __restrict__ b1, float* __restrict__ z,
    float* __restrict__ sum, float* __restrict__ sumsq, int N) {
  __shared__ v2f ldsP[32][PAIR_STRIDE];
  int tid = threadIdx.x;
  stage_w_pairs(ldsP, W1, tid);
  __syncthreads();

  int wave = tid >> 5, lane = tid & 31;
  int l15 = lane & 15;
  int hiK = (lane >> 4) * 2;  // lanes 16-31 hold K = k+2, k+3
  int r0 = blockIdx.x * 128 + wave * 16;
  int rowA = r0 + l15;
  if (rowA >= N) rowA = N - 1;  // clamp: keep EXEC all-1s for WMMA
  const float2* h2 = (const float2*)(h + (long long)rowA * IN_F);

  v8f zero8 = {};
  v8f acc[8];
#pragma unroll
  for (int t = 0; t < 8; ++t) acc[t] = zero8;

#pragma unroll
  for (int kk = 0; kk < IN_F; kk += 4) {
    int kb = kk + hiK;        // even
    int prow = kb >> 1;
    float2 av = h2[prow];
    v2f a;
    a.x = av.x;
    a.y = av.y;
#pragma unroll
    for (int t = 0; t < 8; ++t) {
      v2f b = ldsP[prow][t * 16 + l15];  // single aligned ds_load_b64
      acc[t] = __builtin_amdgcn_wmma_f32_16x16x4_f32(
          false, a, false, b, (short)0, acc[t], false, false);
    }
  }

  // Epilogue: bias + store z, fused per-column BN partial stats.
  // C/D layout: VGPR j holds M=j (lanes 0-15) / M=j+8 (lanes 16-31).
  int hi8 = (lane >> 4) * 8;
#pragma unroll
  for (int t = 0; t < 8; ++t) {
    int c = t * 16 + l15;
    float bias = b1[c];
    float s = 0.0f, q = 0.0f;
#pragma unroll
    for (int j = 0; j < 8; ++j) {
      int row = r0 + hi8 + j;
      if (row < N) {
        float v = acc[t][j] + bias;
        z[(long long)row * OUT_F + c] = v;
        s += v;
        q += v * v;
      }
    }
    // lane L and L+16 hold the same column (rows 0-7 vs 8-15): fold halves
    s += __shfl_xor(s, 16, 32);
    q += __shfl_xor(q, 16, 32);
    if (lane < 16) {
      atomicAdd(&sum[c], s);
      atomicAdd(&sumsq[c], q);
    }
  }
}

__global__ void gin_bnfin(const float* __restrict__ sum,
                          const float* __restrict__ sumsq,
                          const float* __restrict__ gamma,
                          const float* __restrict__ beta,
                          float* __restrict__ scale, float* __restrict__ shift,
                          int N) {
  int c = threadIdx.x;
  float inv = 1.0f / (float)N;
  float mean = sum[c] * inv;
  float var = sumsq[c] * inv - mean * mean;
  float sc = gamma[c] * rsqrtf(var + BN_EPS_F);
  scale[c] = sc;
  shift[c] = beta[c] - mean * sc;
}

// out = relu( relu(z*scale+shift) @ W2 + b2 )   (K=128, two LDS halves)
__global__ __launch_bounds__(256) void gin_gemm2(
    const float* __restrict__ z, const float* __restrict__ W2,
    const float* __restrict__ b2, const float* __restrict__ scale,
    const float* __restrict__ shift, float* __restrict__ out, int N) {
  __shared__ v2f ldsP[32][PAIR_STRIDE];
  __shared__ float ldsS[128];
  __shared__ float ldsT[128];
  int tid = threadIdx.x;
  if (tid < 128) {
    ldsS[tid] = scale[tid];
    ldsT[tid] = shift[tid];
  }

  int wave = tid >> 5, lane = tid & 31;
  int l15 = lane & 15;
  int hiK = (lane >> 4) * 2;
  int r0 = blockIdx.x * 128 + wave * 16;
  int rowA = r0 + l15;
  if (rowA >= N) rowA = N - 1;
  const float2* z2 = (const float2*)(z + (long long)rowA * OUT_F);

  v8f zero8 = {};
  v8f acc[8];
#pragma unroll
  for (int t = 0; t < 8; ++t) acc[t] = zero8;

  for (int half = 0; half < 2; ++half) {
    __syncthreads();  // protects previous-half LDS reads (and ldsS/ldsT)
    stage_w_pairs(ldsP, W2 + half * 64 * OUT_F, tid);
    __syncthreads();

#pragma unroll
    for (int kk = 0; kk < 64; kk += 4) {
      int kb = kk + hiK;      // even
      int prow = kb >> 1;
      int kg = half * 64 + kb;
      float2 av = z2[kg >> 1];
      v2f a;  // fused BatchNorm + ReLU on the A-fragment
      a.x = fmaxf(av.x * ldsS[kg] + ldsT[kg], 0.0f);
      a.y = fmaxf(av.y * ldsS[kg + 1] + ldsT[kg + 1], 0.0f);
#pragma unroll
      for (int t = 0; t < 8; ++t) {
        v2f b = ldsP[prow][t * 16 + l15];
        acc[t] = __builtin_amdgcn_wmma_f32_16x16x4_f32(
            false, a, false, b, (short)0, acc[t], false, false);
      }
    }
  }

  int hi8 = (lane >> 4) * 8;
#pragma unroll
  for (int t = 0; t < 8; ++t) {
    int c = t * 16 + l15;
    float bias = b2[c];
#pragma unroll
    for (int j = 0; j < 8; ++j) {
      int row = r0 + hi8 + j;
      if (row < N)
        out[(long long)row * OUT_F + c] = fmaxf(acc[t][j] + bias, 0.0f);
    }
  }
}

extern "C" void kernel_launch(void* const* d_in, const int* in_sizes, int n_in,
                              void* d_out, int out_size, void* d_ws,
                              size_t ws_size, hipStream_t stream) {
  const float* x = (const float*)d_in[0];
  const long long* ei = (const long long*)d_in[1];  // int64 [2][E]
  const float* W1 = (const float*)d_in[2];
  const float* b1 = (const float*)d_in[3];
  const float* gamma = (const float*)d_in[4];
  const float* beta = (const float*)d_in[5];
  const float* W2 = (const float*)d_in[6];
  const float* b2 = (const float*)d_in[7];

  int N = in_sizes[0] / IN_F;
  int E = in_sizes[1] / 2;

  float* ws = (float*)d_ws;
  float* h = ws;
  float* z = h + (size_t)N * IN_F;
  float* stats = z + (size_t)N * OUT_F;
  float* sum = stats;
  float* sumsq = stats + 128;
  float* scale = stats + 256;
  float* shift = stats + 384;
  float* out = (float*)d_out;

  int total4 = N * IN_F / 4;
  gin_init<<<(total4 + 255) / 256, 256, 0, stream>>>(x, h, stats, total4);

  long long threadsS = (long long)E * 32;
  gin_scatter<<<(unsigned)((threadsS + 255) / 256), 256, 0, stream>>>(
      x, ei, ei + E, h, E);

  int gblocks = (N + 127) / 128;
  gin_gemm1<<<gblocks, 256, 0, stream>>>(h, W1, b1, z, sum, sumsq, N);
  gin_bnfin<<<1, 128, 0, stream>>>(sum, sumsq, gamma, beta, scale, shift, N);
  gin_gemm2<<<gblocks, 256, 0, stream>>>(z, W2, b2, scale, shift, out, N);
}